// GAT_layer_55344948576541
// MI455X (gfx1250) — compile-verified
//
#include <hip/hip_runtime.h>
#include <hip/hip_bf16.h>

#define N_NODES 10000
#define N_EDGES 320000
#define F_INF   128
#define F_OUTF  128
#define H_HEADS 4
#define C_OUT   (H_HEADS * F_OUTF)   // 512
#define STEPS   2
#define NEG_SLOPE 0.2f

typedef float v2f __attribute__((ext_vector_type(2)));
typedef float v8f __attribute__((ext_vector_type(8)));

// ---------------------------------------------------------------------------
// GEMM: h[N,512] = x[N,128] @ W[128,512] using V_WMMA_F32_16X16X4_F32.
// One wave32 per 16x16 output tile. 625 row tiles * 32 col tiles = 20000 tiles,
// 8 waves per 256-thread block -> 2500 blocks. N=10000 is exactly 625*16.
// ---------------------------------------------------------------------------
__global__ __launch_bounds__(256) void gemm_xw(const float* __restrict__ x,
                                               const float* __restrict__ W,
                                               float* __restrict__ h) {
  const int lane = threadIdx.x & 31;
  const int wv   = threadIdx.x >> 5;
  const int tile = blockIdx.x * 8 + wv;          // 0..19999
  const int tiles_n = C_OUT / 16;                // 32
  const int tm = tile / tiles_n;
  const int tn = tile % tiles_n;
  const int m0 = tm * 16, n0 = tn * 16;
  const int half = lane >> 4;                    // K half select (ISA A/B layout)
  const int mr   = lane & 15;                    // row (A) / col (B) within tile

  v8f acc = {};
  const float* arow = x + (size_t)(m0 + mr) * F_INF;
  #pragma unroll 4
  for (int k0 = 0; k0 < F_INF; k0 += 4) {
    const int k = k0 + 2 * half;
    // A 16x4: lanes0-15 hold K={0,1}, lanes16-31 hold K={2,3}
    v2f a; a.x = arow[k]; a.y = arow[k + 1];
    // B 4x16: mirrored layout, row-major W[k][c]
    v2f b; b.x = W[(size_t)k * C_OUT + n0 + mr];
           b.y = W[(size_t)(k + 1) * C_OUT + n0 + mr];
    acc = __builtin_amdgcn_wmma_f32_16x16x4_f32(
        /*neg_a=*/false, a, /*neg_b=*/false, b,
        /*c_mod=*/(short)0, acc, /*reuse_a=*/false, /*reuse_b=*/false);
  }
  // C/D layout: VGPR i -> lanes0-15: M=i, lanes16-31: M=i+8; N = lane&15
  const int nn = lane & 15;
  const int mbase = half * 8;
  #pragma unroll
  for (int i = 0; i < 8; ++i)
    h[(size_t)(m0 + mbase + i) * C_OUT + n0 + nn] = acc[i];
}

// ---------------------------------------------------------------------------
// Per-node attention logits: el[n,h] = <h[n,h,:], attn_l[h,:]>, same for er.
// One block of 128 threads per node; one wave32 per head.
// ---------------------------------------------------------------------------
__global__ __launch_bounds__(128) void node_logits(const float* __restrict__ h,
                                                   const float* __restrict__ attn_l,
                                                   const float* __restrict__ attn_r,
                                                   float* __restrict__ el,
                                                   float* __restrict__ er) {
  const int n = blockIdx.x;
  const int head = threadIdx.x >> 5, lane = threadIdx.x & 31;
  const float* hr = h + (size_t)n * C_OUT + head * F_OUTF;
  const float* al = attn_l + head * F_OUTF;
  const float* ar = attn_r + head * F_OUTF;
  float sl = 0.f, sr = 0.f;
  #pragma unroll
  for (int f = lane; f < F_OUTF; f += 32) {
    const float v = hr[f];
    sl += v * al[f];
    sr += v * ar[f];
  }
  #pragma unroll
  for (int off = 16; off; off >>= 1) {
    sl += __shfl_xor(sl, off, 32);
    sr += __shfl_xor(sr, off, 32);
  }
  if (lane == 0) {
    el[n * H_HEADS + head] = sl;
    er[n * H_HEADS + head] = sr;
  }
}

// ---------------------------------------------------------------------------
// CSR build (once per launch; graph topology is constant across the 2 steps)
// ---------------------------------------------------------------------------
__global__ void edge_count(const int* __restrict__ dst, int* __restrict__ cnt) {
  const int e = blockIdx.x * blockDim.x + threadIdx.x;
  if (e < N_EDGES) atomicAdd(&cnt[dst[e]], 1);
}

__global__ __launch_bounds__(1024) void scan_rowptr(const int* __restrict__ cnt,
                                                    int* __restrict__ row_ptr,
                                                    int* __restrict__ cursor) {
  __shared__ int sh[1024];
  __shared__ int carry;
  const int t = threadIdx.x;
  if (t == 0) carry = 0;
  __syncthreads();
  for (int base = 0; base < N_NODES; base += 1024) {
    const int idx = base + t;
    const int v = (idx < N_NODES) ? cnt[idx] : 0;
    sh[t] = v;
    __syncthreads();
    for (int off = 1; off < 1024; off <<= 1) {   // Hillis-Steele inclusive scan
      const int u = (t >= off) ? sh[t - off] : 0;
      __syncthreads();
      sh[t] += u;
      __syncthreads();
    }
    const int excl = sh[t] - v + carry;
    if (idx < N_NODES) { row_ptr[idx] = excl; cursor[idx] = excl; }
    __syncthreads();
    if (t == 1023) carry += sh[1023];
    __syncthreads();
  }
  if (t == 0) row_ptr[N_NODES] = carry;
}

__global__ void edge_fill(const int* __restrict__ dst, int* __restrict__ cursor,
                          int* __restrict__ edge_idx) {
  const int e = blockIdx.x * blockDim.x + threadIdx.x;
  if (e < N_EDGES) {
    const int p = atomicAdd(&cursor[dst[e]], 1);
    edge_idx[p] = e;
  }
}

// ---------------------------------------------------------------------------
// Fused edge-softmax + aggregation + bias + head-mean + residual.
// Block = 128 threads = 4 wave32 (one head each), one block per dst node.
// Gather-based via CSR: no atomics, h rows are L2-resident (20.5MB << 192MB).
// ---------------------------------------------------------------------------
__global__ __launch_bounds__(128) void gat_aggregate(
    const float* __restrict__ h, const float* __restrict__ el,
    const float* __restrict__ er, const int* __restrict__ src,
    const int* __restrict__ row_ptr, const int* __restrict__ edge_idx,
    const float* __restrict__ bias, const float* __restrict__ alpha_p,
    const float* __restrict__ x0, float* __restrict__ xout) {
  __shared__ float red[H_HEADS * F_OUTF];
  const int n = blockIdx.x;
  const int head = threadIdx.x >> 5, lane = threadIdx.x & 31;
  const int rs = row_ptr[n], re = row_ptr[n + 1];
  const float erd = er[n * H_HEADS + head];

  // Phase 1: segment max (lane-parallel over edges, wave reduce)
  float mx = -INFINITY;
  for (int i = rs + lane; i < re; i += 32) {
    const int e = edge_idx[i];
    float v = el[src[e] * H_HEADS + head] + erd;
    v = v > 0.f ? v : NEG_SLOPE * v;
    mx = fmaxf(mx, v);
  }
  #pragma unroll
  for (int off = 16; off; off >>= 1) mx = fmaxf(mx, __shfl_xor(mx, off, 32));

  // Phase 2: softmax denominator
  float sum = 0.f;
  for (int i = rs + lane; i < re; i += 32) {
    const int e = edge_idx[i];
    float v = el[src[e] * H_HEADS + head] + erd;
    v = v > 0.f ? v : NEG_SLOPE * v;
    sum += __expf(v - mx);
  }
  #pragma unroll
  for (int off = 16; off; off >>= 1) sum += __shfl_xor(sum, off, 32);
  const float inv = (sum != 0.f) ? 1.f / sum : 0.f;

  // Phase 3: weighted feature aggregation (float4 per lane = 128 features/wave)
  float4 acc = make_float4(0.f, 0.f, 0.f, 0.f);
  for (int i = rs; i < re; ++i) {
    const int e = edge_idx[i];
    const int s = src[e];
    float v = el[s * H_HEADS + head] + erd;
    v = v > 0.f ? v : NEG_SLOPE * v;
    const float a = __expf(v - mx) * inv;
    const float4 hv = *(const float4*)(h + (size_t)s * C_OUT + head * F_OUTF + lane * 4);
    acc.x += a * hv.x; acc.y += a * hv.y; acc.z += a * hv.z; acc.w += a * hv.w;
  }
  const float4 bv = *(const float4*)(bias + head * F_OUTF + lane * 4);
  acc.x += bv.x; acc.y += bv.y; acc.z += bv.z; acc.w += bv.w;
  *(float4*)(red + head * F_OUTF + lane * 4) = acc;
  __syncthreads();

  // Head-mean + residual blend
  const int f = threadIdx.x;
  const float s = 0.25f * (red[f] + red[F_OUTF + f] +
                           red[2 * F_OUTF + f] + red[3 * F_OUTF + f]);
  const float al = alpha_p[0];
  xout[(size_t)n * F_OUTF + f] = (1.f - al) * s + al * x0[(size_t)n * F_OUTF + f];
}

// ---------------------------------------------------------------------------
extern "C" void kernel_launch(void* const* d_in, const int* in_sizes, int n_in,
                              void* d_out, int out_size, void* d_ws, size_t ws_size,
                              hipStream_t stream) {
  const float* x      = (const float*)d_in[0];
  const float* x0     = (const float*)d_in[1];
  const int*   src    = (const int*)d_in[2];
  const int*   dst    = (const int*)d_in[3];
  const float* alpha  = (const float*)d_in[4];
  const float* W      = (const float*)d_in[5];
  const float* attn_l = (const float*)d_in[6];
  const float* attn_r = (const float*)d_in[7];
  const float* bias   = (const float*)d_in[8];

  char* ws = (char*)d_ws;
  size_t off = 0;
  auto alloc = [&](size_t bytes) -> char* {
    char* p = ws + off;
    off = (off + bytes + 255) & ~(size_t)255;
    return p;
  };
  float* h        = (float*)alloc((size_t)N_NODES * C_OUT * sizeof(float));   // 20.5 MB
  float* el       = (float*)alloc((size_t)N_NODES * H_HEADS * sizeof(float));
  float* er       = (float*)alloc((size_t)N_NODES * H_HEADS * sizeof(float));
  float* xbuf     = (float*)alloc((size_t)N_NODES * F_INF * sizeof(float));   // 5.1 MB
  int*   cnt      = (int*)alloc((size_t)N_NODES * sizeof(int));
  int*   row_ptr  = (int*)alloc((size_t)(N_NODES + 1) * sizeof(int));
  int*   cursor   = (int*)alloc((size_t)N_NODES * sizeof(int));
  int*   edge_idx = (int*)alloc((size_t)N_EDGES * sizeof(int));               // 1.3 MB

  // One-time CSR build (topology identical across both steps)
  hipMemsetAsync(cnt, 0, (size_t)N_NODES * sizeof(int), stream);
  edge_count<<<(N_EDGES + 255) / 256, 256, 0, stream>>>(dst, cnt);
  scan_rowptr<<<1, 1024, 0, stream>>>(cnt, row_ptr, cursor);
  edge_fill<<<(N_EDGES + 255) / 256, 256, 0, stream>>>(dst, cursor, edge_idx);

  const float* xin = x;
  for (int step = 0; step < STEPS; ++step) {
    gemm_xw<<<2500, 256, 0, stream>>>(xin, W, h);
    node_logits<<<N_NODES, 128, 0, stream>>>(h, attn_l, attn_r, el, er);
    float* xo = (step == STEPS - 1) ? (float*)d_out : xbuf;
    gat_aggregate<<<N_NODES, 128, 0, stream>>>(h, el, er, src, row_ptr, edge_idx,
                                               bias, alpha, x0, xo);
    xin = xbuf;
  }
}